// TGCNCell_60352880443529
// MI455X (gfx1250) — compile-verified
//
#include <hip/hip_runtime.h>
#include <cstdint>
#include <cstddef>

typedef __attribute__((ext_vector_type(2))) float v2f;
typedef __attribute__((ext_vector_type(8))) float v8f;

#if __has_builtin(__builtin_amdgcn_global_load_async_to_lds_b128) && \
    __has_builtin(__builtin_amdgcn_s_wait_asynccnt)
#define TGCN_ASYNC_LDS 1
// exact pointee type from the builtin prototype: GCC-vector int4
typedef int v4i_gcc __attribute__((vector_size(16)));
typedef __attribute__((address_space(1))) v4i_gcc* gptr_v4i;
typedef __attribute__((address_space(3))) v4i_gcc* lptr_v4i;
#endif

__device__ __forceinline__ float sigmoidf_(float x) {
    return 1.0f / (1.0f + __expf(-x));
}

// ---------------- degree / normalization ----------------
__global__ void deg_init_kernel(float* deg, int n) {
    int i = blockIdx.x * blockDim.x + threadIdx.x;
    if (i < n) deg[i] = 1.0f;               // self-loop contributes 1
}

__global__ void deg_count_kernel(const int* __restrict__ dst, float* deg, int e) {
    int i = blockIdx.x * blockDim.x + threadIdx.x;
    if (i < e) atomicAdd(&deg[dst[i]], 1.0f);
}

__global__ void deg_rsqrt_kernel(float* deg, int n) {
    int i = blockIdx.x * blockDim.x + threadIdx.x;
    if (i < n) deg[i] = rsqrtf(deg[i]);     // deg >= 1 always
}

// ---------------- W pre-interleave: Wp[((k/2)*outc + c)*2 + (k&1)] = W[k][c] --
// Makes each WMMA B-fragment (W[k][c], W[k+1][c]) an adjacent 8-byte pair so the
// GEMM can fetch it with a single ds_load_b64 (and async-DMA it linearly).
__global__ void interleave_w_kernel(const float* __restrict__ W,
                                    float* __restrict__ Wp,
                                    int total, int outc) {
    int i = blockIdx.x * blockDim.x + threadIdx.x;
    if (i >= total) return;
    const int k = i / outc, c = i - k * outc;
    Wp[((k >> 1) * outc + c) * 2 + (k & 1)] = W[i];
}

// ---------------- fp32 WMMA GEMM + fused self-loop epilogue ----------------
// outH[N,OUTC] = [xa|xb][N,128] @ W ;  outP = outH * dinv[row]^2
// One wave computes a 16 x OUTC tile with V_WMMA_F32_16X16X4_F32 (full fp32).
template<int OUTC>
__global__ __launch_bounds__(256) void gemm128_wmma_kernel(
    const float* __restrict__ xa, int sa,     // first 64 features, row stride sa
    const float* __restrict__ xb, int sb,     // last 64 features, row stride sb
    float* __restrict__ Wp,                   // pair-interleaved [64][OUTC][2]
    const float* __restrict__ dinv,
    float* __restrict__ outH,                 // [nRows, OUTC]
    float* __restrict__ outP,                 // [nRows, OUTC] self-loop init
    int nRows)
{
    __shared__ float sW[128 * OUTC];
#ifdef TGCN_ASYNC_LDS
    {   // async DMA (ASYNCcnt-tracked) of the pre-interleaved W into LDS
        const int bytesTot = 128 * OUTC * 4;
        for (int off = (int)threadIdx.x * 16; off < bytesTot; off += 256 * 16) {
            __builtin_amdgcn_global_load_async_to_lds_b128(
                (gptr_v4i)((char*)Wp + off),
                (lptr_v4i)((char*)sW + off),
                0, 0);
        }
        __builtin_amdgcn_s_wait_asynccnt(0);
    }
#else
    for (int i = threadIdx.x; i < 128 * OUTC; i += 256) sW[i] = Wp[i];
#endif
    __syncthreads();

    constexpr int NT = OUTC / 16;
    const int wave = threadIdx.x >> 5;
    const int lane = threadIdx.x & 31;
    const int tile = blockIdx.x * 8 + wave;
    const int nTiles = (nRows + 15) >> 4;
    if (tile >= nTiles) return;               // wave-uniform: EXEC stays all-ones

    const int m0 = tile << 4;
    int mRow = m0 + (lane & 15);
    if (mRow >= nRows) mRow = nRows - 1;      // clamp loads, guard stores below
    const int kHalf = (lane >> 4) << 1;       // 0 for lanes 0-15, 2 for 16-31
    const int col   = lane & 15;

    const float* rowA = xa + (size_t)mRow * sa;
    const float* rowB = xb + (size_t)mRow * sb;

    v8f acc[NT] = {};
    for (int kk = 0; kk < 128; kk += 4) {
        const int kA = kk + kHalf;            // even; never straddles the 64 split
        v2f a;
        if (kA < 64) { a.x = rowA[kA];      a.y = rowA[kA + 1];  }
        else         { a.x = rowB[kA - 64]; a.y = rowB[kA - 63]; }
        const int pRow = (kA >> 1) * OUTC;    // interleaved pair-row base
        #pragma unroll
        for (int t = 0; t < NT; ++t) {
            const v2f b = *(const v2f*)&sW[(pRow + t * 16 + col) * 2];
            acc[t] = __builtin_amdgcn_wmma_f32_16x16x4_f32(
                false, a, false, b, (short)0, acc[t], false, false);
        }
    }

    const int rBase = m0 + ((lane >> 4) << 3);   // lanes 16-31 hold rows +8
    #pragma unroll
    for (int t = 0; t < NT; ++t) {
        #pragma unroll
        for (int r = 0; r < 8; ++r) {
            const int row = rBase + r;
            if (row < nRows) {
                const float v  = acc[t][r];
                const float di = dinv[row];
                outH[(size_t)row * OUTC + t * 16 + col] = v;
                outP[(size_t)row * OUTC + t * 16 + col] = v * di * di;
            }
        }
    }
}

// ---------------- edge scatter: out[dst] += h[src] * dinv[src]*dinv[dst] ------
template<int C>
__global__ void edge_prop_kernel(const int* __restrict__ src,
                                 const int* __restrict__ dst,
                                 const float* __restrict__ dinv,
                                 const float* __restrict__ h,
                                 float* __restrict__ out, int e) {
    constexpr int TPE = C / 4;                 // lanes per edge, float4 each
    const long long tid = (long long)blockIdx.x * blockDim.x + threadIdx.x;
    const int ei = (int)(tid / TPE);
    if (ei >= e) return;
    const int c0 = ((int)(tid % TPE)) * 4;
    const int s = src[ei], d = dst[ei];
    const float nrm = dinv[s] * dinv[d];
    const float4 v = *reinterpret_cast<const float4*>(h + (size_t)s * C + c0);
    float* o = out + (size_t)d * C + c0;
    atomicAdd(o + 0, v.x * nrm);
    atomicAdd(o + 1, v.y * nrm);
    atomicAdd(o + 2, v.z * nrm);
    atomicAdd(o + 3, v.w * nrm);
}

// ---------------- c_input = [x | r * hidden], r from flat-split trick ---------
__global__ void build_cinput_kernel(const float* __restrict__ x,
                                    const float* __restrict__ hidden,
                                    const float* __restrict__ out1,
                                    const float* __restrict__ b1,
                                    float* __restrict__ cin, int nh) {
    int j = blockIdx.x * blockDim.x + threadIdx.x;
    if (j >= nh) return;
    const int n = j >> 6, c = j & 63;
    const float r = sigmoidf_(out1[j] + b1[j & 127]);   // r_flat[j] = ru_flat[j]
    cin[(size_t)n * 128 + c]      = x[j];
    cin[(size_t)n * 128 + 64 + c] = r * hidden[j];
}

// ---------------- h' = u*h + (1-u)*tanh(out2 + b2), in place on d_out ---------
__global__ void finalize_kernel(const float* __restrict__ hidden,
                                const float* __restrict__ out1,
                                const float* __restrict__ b1,
                                const float* __restrict__ b2,
                                float* __restrict__ out, int nh) {
    int j = blockIdx.x * blockDim.x + threadIdx.x;
    if (j >= nh) return;
    const float u  = sigmoidf_(out1[nh + j] + b1[j & 127]);  // u_flat[j]=ru_flat[nh+j]
    const float cv = tanhf(out[j] + b2[j & 63]);
    out[j] = u * hidden[j] + (1.0f - u) * cv;
}

extern "C" void kernel_launch(void* const* d_in, const int* in_sizes, int n_in,
                              void* d_out, int out_size, void* d_ws, size_t ws_size,
                              hipStream_t stream) {
    const float* x      = (const float*)d_in[0];
    const float* hidden = (const float*)d_in[1];
    const int*   ei     = (const int*)d_in[2];
    const float* W1     = (const float*)d_in[3];   // [128,128]
    const float* b1     = (const float*)d_in[4];   // [128]
    const float* W2     = (const float*)d_in[5];   // [128,64]
    const float* b2     = (const float*)d_in[6];   // [64]

    const int CIN = 64, H = 64;
    const int N  = in_sizes[0] / CIN;
    const int E  = in_sizes[2] / 2;
    const int NH = N * H;

    const int* src = ei;          // edge_index[0]
    const int* dst = ei + E;      // edge_index[1]

    char* ws = (char*)d_ws;
    size_t off = 0;
    auto carve = [&](size_t bytes) -> void* {
        void* p = ws + off;
        off = (off + bytes + 255) & ~(size_t)255;
        return p;
    };
    float* dinv = (float*)carve((size_t)N * 4);
    float* h1   = (float*)carve((size_t)N * 128 * 4);   // reused as c_input
    float* out1 = (float*)carve((size_t)N * 128 * 4);   // kept: u read at the end
    float* h2   = (float*)carve((size_t)N * 64 * 4);
    float* Wp1  = (float*)carve((size_t)128 * 128 * 4); // interleaved W1
    float* Wp2  = (float*)carve((size_t)128 * 64 * 4);  // interleaved W2
    (void)ws_size; (void)n_in; (void)out_size;

    const int B = 256;
    // 0) pre-interleave weights for ds_load_b64-friendly LDS layout
    interleave_w_kernel<<<(128 * 128 + B - 1) / B, B, 0, stream>>>(W1, Wp1, 128 * 128, 128);
    interleave_w_kernel<<<(128 * 64  + B - 1) / B, B, 0, stream>>>(W2, Wp2, 128 * 64,  64);

    // 1) symmetric normalization
    deg_init_kernel <<<(N + B - 1) / B, B, 0, stream>>>(dinv, N);
    deg_count_kernel<<<(E + B - 1) / B, B, 0, stream>>>(dst, dinv, E);
    deg_rsqrt_kernel<<<(N + B - 1) / B, B, 0, stream>>>(dinv, N);

    const int nTiles  = (N + 15) / 16;
    const int gBlocks = (nTiles + 7) / 8;

    // 2) GCNConv1: h1 = [x|hidden] @ W1 ; out1 pre-seeded with self-loop term
    gemm128_wmma_kernel<128><<<gBlocks, B, 0, stream>>>(
        x, CIN, hidden, H, Wp1, dinv, h1, out1, N);
    {
        long long tot = (long long)E * 32;
        edge_prop_kernel<128><<<(unsigned)((tot + B - 1) / B), B, 0, stream>>>(
            src, dst, dinv, h1, out1, E);
    }

    // 3) c_input = [x | r*hidden] into h1 (h1 dead after propagation)
    build_cinput_kernel<<<(NH + B - 1) / B, B, 0, stream>>>(x, hidden, out1, b1, h1, NH);

    // 4) GCNConv2: h2 = c_input @ W2 ; d_out pre-seeded with self-loop term
    gemm128_wmma_kernel<64><<<gBlocks, B, 0, stream>>>(
        h1, 128, h1 + 64, 128, Wp2, dinv, h2, (float*)d_out, N);
    {
        long long tot = (long long)E * 16;
        edge_prop_kernel<64><<<(unsigned)((tot + B - 1) / B), B, 0, stream>>>(
            src, dst, dinv, h2, (float*)d_out, E);
    }

    // 5) gate + blend, in place
    finalize_kernel<<<(NH + B - 1) / B, B, 0, stream>>>(hidden, out1, b1, b2,
                                                        (float*)d_out, NH);
}